// PGCA_54769422959169
// MI455X (gfx1250) — compile-verified
//
#include <hip/hip_runtime.h>

// ---------------------------------------------------------------------------
// PGCA forward for MI455X (gfx1250): L2-resident atomic SpMM + fp32 WMMA gates
// ---------------------------------------------------------------------------

typedef float v2f __attribute__((ext_vector_type(2)));
typedef float v8f __attribute__((ext_vector_type(8)));

#define N_NODE  100000
#define N_USER  40000
#define N_PRICE 100
#define EMB     128

// ----------------------------- zero fill -----------------------------------
__global__ void pgca_zero(float* __restrict__ p, long n) {
    long i = (long)blockIdx.x * blockDim.x + threadIdx.x;
    if (i < n) p[i] = 0.0f;
}

// --------------------------- COO SpMM (atomic) -----------------------------
// One thread handles (edge, 4-float feature chunk). Optional gather index
// (src row = gidx[col]) and optional row-sum denominator accumulation.
__global__ void pgca_spmm(const int* __restrict__ rows, const int* __restrict__ cols,
                          const float* __restrict__ vals, const float* __restrict__ X,
                          const int* __restrict__ gidx,
                          float* __restrict__ out, float* __restrict__ den, int nnz) {
    long tid = (long)blockIdx.x * blockDim.x + threadIdx.x;
    long e = tid >> 5;                 // 32 chunks of 4 floats = 128 features
    int  q = (int)(tid & 31);
    if (e >= nnz) return;
    int   r = rows[e];
    int   c = cols[e];
    float v = vals[e];
    int src = gidx ? gidx[c] : c;
    const float4 x = ((const float4*)(X + (long)src * EMB))[q];
    float* o = out + (long)r * EMB + q * 4;
    atomicAdd(o + 0, v * x.x);
    atomicAdd(o + 1, v * x.y);
    atomicAdd(o + 2, v * x.z);
    atomicAdd(o + 3, v * x.w);
    if (den && q == 0) atomicAdd(den + r, v);
}

// ------------------- intra_gate normalization / scaling --------------------
// out[n,f] (or += ) = scale * (num[n,f] / (den[n] + 1e-8)) * matv[n]
__global__ void pgca_alpha_scale(const float* __restrict__ num, const float* __restrict__ den,
                                 const float* __restrict__ matv, const float* __restrict__ scale_ptr,
                                 float* __restrict__ out, int accumulate, long nrows) {
    long i = (long)blockIdx.x * blockDim.x + threadIdx.x;
    long n = nrows * EMB;
    if (i >= n) return;
    long r = i >> 7;
    float s = scale_ptr ? *scale_ptr : 1.0f;
    float v = s * (num[i] / (den[r] + 1e-8f)) * matv[r];
    if (accumulate) out[i] += v; else out[i] = v;
}

// ------------------- fold Wa[:, :128] + Wb, ba + bb ------------------------
__global__ void pgca_prep_w(const float* __restrict__ Wa, const float* __restrict__ Wb,
                            const float* __restrict__ ba, const float* __restrict__ bb,
                            float* __restrict__ Wc, float* __restrict__ bc) {
    int i = blockIdx.x * blockDim.x + threadIdx.x;
    if (i < EMB * EMB) {
        int o = i >> 7, k = i & 127;
        Wc[i] = Wa[o * (2 * EMB) + k] + Wb[i];
    }
    if (i < EMB) bc[i] = ba[i] + bb[i];
}

// ------------------------- WMMA gate GEMM kernel ---------------------------
// Per wave: 16-row strip, full 128-col output (8 tiles of 16), accumulating
// up to 3 sources with V_WMMA_F32_16X16X4_F32.
//   A-frag (16x4 f32): lane l<16 -> row l, K = k0..k0+1 ; lane l>=16 -> K+2
//   B-frag (4x16 f32): B[k][n] = W[n][k]  (same half-wave K split)
//   C/D  (16x16 f32) : VGPR j, lane l: (m = j + 8*(l>=16), n = l%16)
__device__ __forceinline__ void pgca_wmma_src(const float* __restrict__ E,
                                              const float* __restrict__ W, int ldw,
                                              long aBase, int hi, int l16, v8f acc[8]) {
    for (int k0 = 0; k0 < EMB; k0 += 4) {
        const int kk = k0 + 2 * hi;
        const float2 av = *(const float2*)(E + aBase + kk);
        v2f a; a.x = av.x; a.y = av.y;
#pragma unroll
        for (int t = 0; t < 8; ++t) {
            const int ocol = t * 16 + l16;          // output feature = W row
            const float2 bv = *(const float2*)(W + ocol * ldw + kk);
            v2f b; b.x = bv.x; b.y = bv.y;
            acc[t] = __builtin_amdgcn_wmma_f32_16x16x4_f32(
                false, a, false, b, (short)0, acc[t], false, false);
        }
    }
}

// mode 0: out = e1 + g*e2 (+ ADD)          (item / price inter_gate)
// mode 1: out = g*e1 + (1-g)*e2            (user combine)
__global__ __launch_bounds__(256)
void pgca_gate_wmma(const float* __restrict__ E1, const float* __restrict__ E2,
                    const float* __restrict__ E3,
                    const float* __restrict__ W1, int ld1,
                    const float* __restrict__ W2, int ld2,
                    const float* __restrict__ W3, int ld3,
                    const float* __restrict__ bias, const float* __restrict__ ADD,
                    float* __restrict__ OUT, int N, int mode) {
    const int lane  = threadIdx.x & 31;
    const int wave  = threadIdx.x >> 5;
    const int strip = blockIdx.x * 8 + wave;
    const int row0  = strip * 16;
    if (row0 >= N) return;                 // wave-uniform exit; EXEC stays full
    const int hi  = lane >> 4;
    const int l16 = lane & 15;

    int arow = row0 + l16;
    if (arow >= N) arow = N - 1;           // clamp loads for partial tail tile
    const long aBase = (long)arow * EMB;

    v8f acc[8];
#pragma unroll
    for (int t = 0; t < 8; ++t) acc[t] = (v8f){0, 0, 0, 0, 0, 0, 0, 0};

    pgca_wmma_src(E1, W1, ld1, aBase, hi, l16, acc);
    pgca_wmma_src(E2, W2, ld2, aBase, hi, l16, acc);
    if (E3) pgca_wmma_src(E3, W3, ld3, aBase, hi, l16, acc);

#pragma unroll
    for (int t = 0; t < 8; ++t) {
        const int col = t * 16 + l16;
        const float b = bias[col];
#pragma unroll
        for (int j = 0; j < 8; ++j) {
            const int m   = j + 8 * hi;
            const int row = row0 + m;
            if (row >= N) continue;        // predicated store only (no WMMA after)
            const long idx = (long)row * EMB + col;
            const float pre = acc[t][j] + b;
            const float g = 1.0f / (1.0f + __expf(-pre));
            float o;
            const float e1 = E1[idx];
            const float e2 = E2[idx];
            if (mode == 0) {
                o = e1 + g * e2;
                if (ADD) o += ADD[idx];
            } else {
                o = g * e1 + (1.0f - g) * e2;
            }
            OUT[idx] = o;
        }
    }
}

// ---------------------------------------------------------------------------
extern "C" void kernel_launch(void* const* d_in, const int* in_sizes, int n_in,
                              void* d_out, int out_size, void* d_ws, size_t ws_size,
                              hipStream_t stream) {
    const float* embedding = (const float*)d_in[0];
    const float* pri_emb   = (const float*)d_in[1];
    const float* user_emb  = (const float*)d_in[2];
    const int*   adj_rows  = (const int*)d_in[3];
    const int*   adj_cols  = (const int*)d_in[4];
    const float* adj_vals  = (const float*)d_in[5];
    const int*   vp_rows   = (const int*)d_in[6];
    const int*   vp_cols   = (const int*)d_in[7];
    const float* vp_vals   = (const float*)d_in[8];
    const int*   vu_rows   = (const int*)d_in[9];
    const int*   vu_cols   = (const int*)d_in[10];
    const float* vu_vals   = (const float*)d_in[11];
    const int*   pv_rows   = (const int*)d_in[12];
    const int*   pv_cols   = (const int*)d_in[13];
    const float* pv_vals   = (const float*)d_in[14];
    const int*   uv_rows   = (const int*)d_in[15];
    const int*   uv_cols   = (const int*)d_in[16];
    const float* uv_vals   = (const float*)d_in[17];
    const int*   ip_idx    = (const int*)d_in[18];
    const float* mat_vu    = (const float*)d_in[19];
    const float* mat_pv    = (const float*)d_in[20];
    const float* mat_uv    = (const float*)d_in[21];
    const float* W_aogi    = (const float*)d_in[22];
    const float* b_aogi    = (const float*)d_in[23];
    const float* W_bgi1    = (const float*)d_in[24];
    const float* b_bgi1    = (const float*)d_in[25];
    const float* W_aogp    = (const float*)d_in[26];
    const float* b_aogp    = (const float*)d_in[27];
    const float* W_bgp1    = (const float*)d_in[28];
    const float* b_bgp1    = (const float*)d_in[29];
    const float* W_user    = (const float*)d_in[30];
    const float* b_user    = (const float*)d_in[31];
    const float* user_lam  = (const float*)d_in[32];

    const int nE_adj = in_sizes[3];
    const int nE_vp  = in_sizes[6];
    const int nE_vu  = in_sizes[9];
    const int nE_pv  = in_sizes[12];
    const int nE_uv  = in_sizes[15];

    float* out_item  = (float*)d_out;                       // 100000 x 128
    float* out_price = out_item + (size_t)N_NODE * EMB;     //    100 x 128
    float* out_user  = out_price + (size_t)N_PRICE * EMB;   //  40000 x 128

    // ---- workspace layout ----
    const size_t S_NODE = (size_t)N_NODE * EMB;
    float* ws      = (float*)d_ws;
    float* bufA    = ws;                       // vp_spmm -> e_price_user
    float* bufB    = bufA + S_NODE;            // adj_spmm(+lambda*intra) -> uv_num
    float* bufC    = bufB + S_NODE;            // vu_num  -> e_item_user
    float* den1    = bufC + S_NODE;            // vu_den  -> uv_den (N_NODE)
    float* pvNum   = den1 + N_NODE;
    float* pvDen   = pvNum + (size_t)N_PRICE * EMB;
    float* priceE2 = pvDen + N_PRICE;
    float* W1item  = priceE2 + (size_t)N_PRICE * EMB;
    float* biasIt  = W1item + EMB * EMB;
    float* W1price = biasIt + EMB;
    float* biasPr  = W1price + EMB * EMB;

    const int TB = 256;
    auto blocks1d = [](long n, int tb) { return (unsigned)((n + tb - 1) / tb); };
    auto spmm_blocks = [&](int nnz) { return blocks1d((long)nnz * 32, TB); };
    auto gate_blocks = [](int N) { return (unsigned)(((N + 15) / 16 + 7) / 8); };

    // ================= Phase A: item =================
    const long zeroN = 3 * (long)S_NODE + N_NODE;
    pgca_zero<<<blocks1d(zeroN, TB), TB, 0, stream>>>(bufA, zeroN);
    pgca_spmm<<<spmm_blocks(nE_vp), TB, 0, stream>>>(vp_rows, vp_cols, vp_vals, pri_emb,
                                                     nullptr, bufA, nullptr, nE_vp);
    pgca_spmm<<<spmm_blocks(nE_adj), TB, 0, stream>>>(adj_rows, adj_cols, adj_vals, embedding,
                                                      nullptr, bufB, nullptr, nE_adj);
    pgca_spmm<<<spmm_blocks(nE_vu), TB, 0, stream>>>(vu_rows, vu_cols, vu_vals, user_emb,
                                                     nullptr, bufC, den1, nE_vu);
    pgca_prep_w<<<blocks1d(EMB * EMB, TB), TB, 0, stream>>>(W_aogi, W_bgi1, b_aogi, b_bgi1,
                                                            W1item, biasIt);
    // bufB += user_lambda * (vu_num / vu_den) * mat_vu
    pgca_alpha_scale<<<blocks1d((long)N_NODE * EMB, TB), TB, 0, stream>>>(
        bufC, den1, mat_vu, user_lam, bufB, 1, N_NODE);
    pgca_gate_wmma<<<gate_blocks(N_NODE), TB, 0, stream>>>(
        embedding, bufA, nullptr,
        W1item, EMB, W_aogi + EMB, 2 * EMB, nullptr, 0,
        biasIt, bufB, out_item, N_NODE, 0);

    // ================= Phase B: price =================
    const long zeroP = (long)N_PRICE * EMB + N_PRICE;
    pgca_zero<<<blocks1d(zeroP, TB), TB, 0, stream>>>(pvNum, zeroP);
    pgca_spmm<<<spmm_blocks(nE_pv), TB, 0, stream>>>(pv_rows, pv_cols, pv_vals, embedding,
                                                     nullptr, pvNum, pvDen, nE_pv);
    pgca_prep_w<<<blocks1d(EMB * EMB, TB), TB, 0, stream>>>(W_aogp, W_bgp1, b_aogp, b_bgp1,
                                                            W1price, biasPr);
    pgca_alpha_scale<<<blocks1d((long)N_PRICE * EMB, TB), TB, 0, stream>>>(
        pvNum, pvDen, mat_pv, nullptr, priceE2, 0, N_PRICE);
    pgca_gate_wmma<<<gate_blocks(N_PRICE), TB, 0, stream>>>(
        pri_emb, priceE2, nullptr,
        W1price, EMB, W_aogp + EMB, 2 * EMB, nullptr, 0,
        biasPr, nullptr, out_price, N_PRICE, 0);

    // ================= Phase C: user =================
    pgca_zero<<<blocks1d(zeroN, TB), TB, 0, stream>>>(bufA, zeroN);
    // e_price_user = uv @ pri_emb[item_price_indices]
    pgca_spmm<<<spmm_blocks(nE_uv), TB, 0, stream>>>(uv_rows, uv_cols, uv_vals, pri_emb,
                                                     ip_idx, bufA, nullptr, nE_uv);
    // uv_num = uv @ embedding ; uv_den = row sums
    pgca_spmm<<<spmm_blocks(nE_uv), TB, 0, stream>>>(uv_rows, uv_cols, uv_vals, embedding,
                                                     nullptr, bufB, den1, nE_uv);
    // e_item_user = (uv_num / uv_den) * mat_uv
    pgca_alpha_scale<<<blocks1d((long)N_USER * EMB, TB), TB, 0, stream>>>(
        bufB, den1, mat_uv, nullptr, bufC, 0, N_USER);
    // user = g*user_emb + (1-g)*e_item_user ;
    // g = sigmoid([user_emb, e_item_user, e_price_user] @ W_user^T + b_user)
    pgca_gate_wmma<<<gate_blocks(N_USER), TB, 0, stream>>>(
        user_emb, bufC, bufA,
        W_user, 3 * EMB, W_user + EMB, 3 * EMB, W_user + 2 * EMB, 3 * EMB,
        b_user, nullptr, out_user, N_USER, 1);
}